// FeatureExtractionComponent_84164179132494
// MI455X (gfx1250) — compile-verified
//
#include <hip/hip_runtime.h>

#define NPTS 2048
#define NB   8
#define KNN  16
#define EPSF 1e-5f
#define BIGF 3.0e38f

typedef __attribute__((ext_vector_type(2))) float v2f;
typedef __attribute__((ext_vector_type(8))) float v8f;
typedef __attribute__((ext_vector_type(4))) unsigned int u32x4;
typedef __attribute__((ext_vector_type(4))) int i32x4;
typedef __attribute__((ext_vector_type(8))) int i32x8;

// ---------------------------------------------------------------------------
// wave32 argmin reduction (value, index), ties -> lower index (matches top_k)
// ---------------------------------------------------------------------------
__device__ __forceinline__ void argmin_wave(float& v, int& c) {
#pragma unroll
  for (int off = 16; off > 0; off >>= 1) {
    float ov = __shfl_xor(v, off, 32);
    int   oc = __shfl_xor(c, off, 32);
    if (ov < v || (ov == v && oc < c)) { v = ov; c = oc; }
  }
}

// ---------------------------------------------------------------------------
// Tensor Data Mover: 1-D contiguous DMA of `nelem` f32 from global -> LDS.
// D# built per CDNA5 ISA 8.3/8.4: group0 = {count=1, lds_addr, global_addr,
// type=2}; group1 = {data_size=4B, tensor_dim0=nelem, tensor_dim1=1,
// tile_dim0=nelem, tile_dim1/2=0, stride0=nelem}. Groups 2/3 zero (<=2D).
// ---------------------------------------------------------------------------
__device__ __forceinline__ void tdm_load_1d_f32(unsigned lds_off,
                                                const void* gptr,
                                                unsigned nelem) {
  unsigned long long ga = (unsigned long long)(size_t)gptr;
  u32x4 g0;
  g0.x = 1u;                                    // count=1, user mode
  g0.y = lds_off;                               // LDS byte address
  g0.z = (unsigned)(ga & 0xFFFFFFFFu);          // global_addr[31:0]
  g0.w = (unsigned)((ga >> 32) & 0x01FFFFFFu) | (2u << 30);  // [56:32]|type=2
  i32x8 g1;
  g1[0] = (int)(2u << 16);                      // wg_mask=0, data_size=2 (4B)
  g1[1] = (int)((nelem & 0xFFFFu) << 16);       // tensor_dim0[15:0]
  g1[2] = (int)((nelem >> 16) | (1u << 16));    // tensor_dim0[31:16]|dim1=1
  g1[3] = (int)((nelem & 0xFFFFu) << 16);       // dim1[31:16]=0 | tile_dim0
  g1[4] = 0;                                    // tile_dim1=0, tile_dim2=0
  g1[5] = (int)nelem;                           // tensor_dim0_stride[31:0]
  g1[6] = 0;                                    // stride0[47:32]|stride1 lo
  g1[7] = 0;
  i32x4 z4 = {0, 0, 0, 0};
#if defined(__clang_major__) && (__clang_major__ >= 23)
  i32x8 z8 = {0, 0, 0, 0, 0, 0, 0, 0};
  __builtin_amdgcn_tensor_load_to_lds(g0, g1, z4, z4, z8, 0);
#else
  __builtin_amdgcn_tensor_load_to_lds(g0, g1, z4, z4, 0);
#endif
}

// ---------------------------------------------------------------------------
// Kernel 1: pre = relu(bn(w_pre @ x + b_pre))  (BN folded), sq = |p|^2,
// and pass-through x -> out channels 84..86.
// ---------------------------------------------------------------------------
__global__ __launch_bounds__(256) void pre_bn_kernel(
    const float* __restrict__ x,  const float* __restrict__ w_pre,
    const float* __restrict__ b_pre, const float* __restrict__ g_pre,
    const float* __restrict__ be_pre, const float* __restrict__ m_pre,
    const float* __restrict__ v_pre,
    float* __restrict__ preA, float* __restrict__ sq, float* __restrict__ out) {
  int gid = blockIdx.x * 256 + threadIdx.x;
  if (gid >= NB * NPTS) return;
  int b = gid >> 11, n = gid & (NPTS - 1);
  const float* xb = x + b * 3 * NPTS;
  float x0 = xb[n], x1 = xb[NPTS + n], x2 = xb[2 * NPTS + n];
  sq[gid] = x0 * x0 + x1 * x1 + x2 * x2;
#pragma unroll
  for (int o = 0; o < 24; ++o) {
    float p = w_pre[o * 3] * x0 + w_pre[o * 3 + 1] * x1 + w_pre[o * 3 + 2] * x2
              + b_pre[o];
    float a = g_pre[o] * rsqrtf(v_pre[o] + EPSF);
    float h = (p - m_pre[o]) * a + be_pre[o];
    preA[(b * 24 + o) * NPTS + n] = fmaxf(h, 0.0f);
  }
  out[(b * 87 + 84) * NPTS + n] = x0;
  out[(b * 87 + 85) * NPTS + n] = x1;
  out[(b * 87 + 86) * NPTS + n] = x2;
}

// ---------------------------------------------------------------------------
// Kernel 2: kNN. One block = one batch x 16 query rows.
// Gram tiles via V_WMMA_F32_16X16X4_F32 (A = 16 rows x [x,y,z,0]).
// Two 1024-column phases keep the f32 d2 strip at 64KB of LDS; phase-0
// top-16 winners live in registers and are merged during phase-1 selection.
// All per-lane fragment loads are unconditional (cndmask-select, no exec
// diamonds in the hot loop); the self-diagonal fix is a wave-uniform branch.
// ---------------------------------------------------------------------------
__global__ __launch_bounds__(256) void knn_kernel(
    const float* __restrict__ x, const float* __restrict__ sq,
    int* __restrict__ nbr) {
  __shared__ float d2s[16 * 1024];  // 64 KB
  const int b    = blockIdx.x >> 7;        // 128 row-groups per batch
  const int n0   = (blockIdx.x & 127) << 4;
  const int lane = threadIdx.x & 31, wave = threadIdx.x >> 5;
  const int l16  = lane & 15, half = lane >> 4;
  const float* xb  = x  + b * 3 * NPTS;
  const float* sqb = sq + b * NPTS;
  const float* xsel = xb + (half ? 2 * NPTS : 0);  // lanes 0-15: x, 16-31: z

  // A fragment: 16x4 f32; lanes 0-15 hold K=0,1 (x,y), lanes 16-31 K=2,3 (z,0)
  float ax = xsel[n0 + l16];
  float ay = xb[NPTS + n0 + l16];
  v2f af;
  af.x = ax;
  af.y = half ? 0.0f : ay;
  float sqr[8];
#pragma unroll
  for (int r = 0; r < 8; ++r) sqr[r] = sqb[n0 + r + 8 * half];

  // phase-0 winners: lane L holds slot (L&15) of row 2*wave + (L>>4)
  float winV = BIGF;
  int   winC = 1 << 30;
  const int myrow = half;  // 0 or 1 within the wave's row pair

  for (int p = 0; p < 2; ++p) {
    if (p) __syncthreads();          // phase-0 selection done before overwrite
    const int cbase = p << 10;
    for (int t = wave; t < 64; t += 8) {
      const int lc  = t << 4;
      const int col = cbase + lc + l16;
      float bxz = xsel[col];
      float by  = xb[NPTS + col];
      v2f bf;
      bf.x = bxz;
      bf.y = half ? 0.0f : by;
      v8f acc = {};
      acc = __builtin_amdgcn_wmma_f32_16x16x4_f32(
          false, af, false, bf, (short)0, acc, false, false);
      const float sqc = sqb[col];
      float val[8];
#pragma unroll
      for (int r = 0; r < 8; ++r) val[r] = sqr[r] + sqc - 2.0f * acc[r];
      if (cbase + lc == n0) {          // wave-uniform: diagonal tile only
#pragma unroll
        for (int r = 0; r < 8; ++r)
          if (l16 == r + 8 * half) val[r] = BIGF;  // exclude self (idx[...,1:])
      }
#pragma unroll
      for (int r = 0; r < 8; ++r)
        d2s[(r + 8 * half) * 1024 + lc + l16] = val[r];
    }
    __syncthreads();

    // selection: wave handles rows 2*wave and 2*wave+1; scan via b128 LDS
    // loads (col = 4*lane + 128*j -> conflict-light, index-ordered for ties)
    for (int rr = 0; rr < 2; ++rr) {
      const int row = wave * 2 + rr;
      float* drow = d2s + row * 1024;
      for (int s = 0; s < KNN; ++s) {
        float bv = BIGF;
        int   bc = 1 << 30;
#pragma unroll
        for (int j = 0; j < 8; ++j) {
          const int c0 = (j * 32 + lane) * 4;
          const float4 q = *(const float4*)&drow[c0];
#pragma unroll
          for (int e = 0; e < 4; ++e) {
            float v = (e == 0) ? q.x : (e == 1) ? q.y : (e == 2) ? q.z : q.w;
            int  gc = cbase + c0 + e;
            if (v < bv || (v == bv && gc < bc)) { bv = v; bc = gc; }
          }
        }
        if (p == 1) {  // merge register-held phase-0 winner (cols < 1024)
          float wv = (myrow == rr) ? winV : BIGF;
          int   wc = (myrow == rr) ? winC : (1 << 30);
          if (wv < bv || (wv == bv && wc < bc)) { bv = wv; bc = wc; }
        }
        argmin_wave(bv, bc);
        if (p == 0) {
          if (myrow == rr && l16 == s) { winV = bv; winC = bc; }
          if (lane == 0) drow[bc] = BIGF;             // consume in strip
        } else {
          if (lane == 0) nbr[((b * NPTS) + n0 + row) * KNN + s] = bc;
          if (bc >= 1024) {
            if (lane == 0) drow[bc - 1024] = BIGF;    // consume in strip
          } else if (myrow == rr && winC == bc) {
            winV = BIGF;                              // consume winner
          }
        }
      }
    }
  }
}

// ---------------------------------------------------------------------------
// One MLP layer as f32 WMMA chain: h[0:12] = W(12 x kTot) @ Y[inBase..] (x16)
// Y rows laid out as [h3 | h2 | h1 | center | grouped] so each layer's input
// is a contiguous row range (free concat).
// ---------------------------------------------------------------------------
__device__ __forceinline__ void mlp_layer(
    float* __restrict__ Y, const float* __restrict__ Wl, int kTot,
    int inBase, int hBase, const float* g, const float* be, const float* mm,
    const float* vv, bool do_bn, int l16, int half) {
  v8f acc = {};
  for (int kk = 0; kk < kTot; kk += 4) {
    v2f af, bf;
    float a0 = 0.0f, a1 = 0.0f;
    if (l16 < 12) {  // rows 12..15 of A are zero padding
      a0 = Wl[l16 * kTot + kk + 2 * half];
      a1 = Wl[l16 * kTot + kk + 2 * half + 1];
    }
    af.x = a0; af.y = a1;
    bf.x = Y[(inBase + kk + 2 * half)     * 16 + l16];
    bf.y = Y[(inBase + kk + 2 * half + 1) * 16 + l16];
    acc = __builtin_amdgcn_wmma_f32_16x16x4_f32(
        false, af, false, bf, (short)0, acc, false, false);
  }
#pragma unroll
  for (int r = 0; r < 8; ++r) {
    int m = r + 8 * half;
    if (m < 12) {
      float h = acc[r];
      if (do_bn) {
        float a = g[m] * rsqrtf(vv[m] + EPSF);
        h = fmaxf((h - mm[m]) * a + be[m], 0.0f);
      }
      Y[(hBase + m) * 16 + l16] = h;
    }
  }
}

// ---------------------------------------------------------------------------
// Kernel 3: one wave = one point (its 16 neighbors are the GEMM columns).
// Weights are DMA'd into LDS by the Tensor Data Mover (one wave issues three
// 1-D descriptors, waits on TENSORcnt, then the block barrier publishes them).
// ---------------------------------------------------------------------------
__global__ __launch_bounds__(256) void mlp_kernel(
    const float* __restrict__ preA, const int* __restrict__ nbr,
    const float* __restrict__ w1, const float* __restrict__ g1,
    const float* __restrict__ be1, const float* __restrict__ m1,
    const float* __restrict__ v1,
    const float* __restrict__ w2, const float* __restrict__ g2,
    const float* __restrict__ be2, const float* __restrict__ m2,
    const float* __restrict__ v2,
    const float* __restrict__ w3, float* __restrict__ out) {
  __shared__ float Wl[12 * 48 + 12 * 60 + 12 * 72];  // 2160 f
  __shared__ float Yall[8][84 * 16];                 // 42 KB

  if (threadIdx.x == 0) {  // only wave 0 reaches the TDM issues
    unsigned base = (unsigned)(size_t)&Wl[0];        // low 32b = LDS offset
    tdm_load_1d_f32(base,              w1, 12 * 48);
    tdm_load_1d_f32(base + 576 * 4,    w2, 12 * 60);
    tdm_load_1d_f32(base + 1296 * 4,   w3, 12 * 72);
    __builtin_amdgcn_s_wait_tensorcnt(0);
  }
  __syncthreads();

  const int lane = threadIdx.x & 31, wave = threadIdx.x >> 5;
  const int l16 = lane & 15, half = lane >> 4;
  const int gl = blockIdx.x * 8 + wave;
  const int b = gl >> 11, n = gl & (NPTS - 1);
  float* Y = Yall[wave];
  const float* pb = preA + b * 24 * NPTS;

  // rows 36..59 = center (broadcast over k), rows 60..83 = grouped neighbors
  if (!half) {
#pragma unroll
    for (int c = 0; c < 24; ++c) Y[(36 + c) * 16 + l16] = pb[c * NPTS + n];
  } else {
    int j = nbr[(b * NPTS + n) * KNN + l16];
#pragma unroll
    for (int c = 0; c < 24; ++c) Y[(60 + c) * 16 + l16] = pb[c * NPTS + j];
  }

  mlp_layer(Y, Wl,        48, 36, 24, g1, be1, m1, v1, true,  l16, half); // h1
  mlp_layer(Y, Wl + 576,  60, 24, 12, g2, be2, m2, v2, true,  l16, half); // h2
  mlp_layer(Y, Wl + 1296, 72, 12,  0, nullptr, nullptr, nullptr, nullptr,
            false, l16, half);                                            // h3

  // max over the 16 neighbor columns -> out channels 0..83
  for (int r = lane; r < 84; r += 32) {
    float mv = Y[r * 16];
#pragma unroll
    for (int k = 1; k < 16; ++k) mv = fmaxf(mv, Y[r * 16 + k]);
    out[(b * 87 + r) * NPTS + n] = mv;
  }
}

// ---------------------------------------------------------------------------
extern "C" void kernel_launch(void* const* d_in, const int* in_sizes, int n_in,
                              void* d_out, int out_size, void* d_ws,
                              size_t ws_size, hipStream_t stream) {
  (void)in_sizes; (void)n_in; (void)out_size; (void)ws_size;
  const float* x      = (const float*)d_in[0];
  const float* w_pre  = (const float*)d_in[1];
  const float* b_pre  = (const float*)d_in[2];
  const float* g_pre  = (const float*)d_in[3];
  const float* be_pre = (const float*)d_in[4];
  const float* m_pre  = (const float*)d_in[5];
  const float* v_pre  = (const float*)d_in[6];
  const float* w1  = (const float*)d_in[7];
  const float* g1  = (const float*)d_in[8];
  const float* be1 = (const float*)d_in[9];
  const float* m1  = (const float*)d_in[10];
  const float* v1  = (const float*)d_in[11];
  const float* w2  = (const float*)d_in[12];
  const float* g2  = (const float*)d_in[13];
  const float* be2 = (const float*)d_in[14];
  const float* m2  = (const float*)d_in[15];
  const float* v2  = (const float*)d_in[16];
  const float* w3  = (const float*)d_in[17];
  float* out = (float*)d_out;

  // workspace layout: preA (B*24*N f) | sq (B*N f) | idx (B*N*16 i32) ~ 2.6MB
  float* preA = (float*)d_ws;
  float* sq   = preA + NB * 24 * NPTS;
  int*   nbr  = (int*)(sq + NB * NPTS);

  pre_bn_kernel<<<(NB * NPTS + 255) / 256, 256, 0, stream>>>(
      x, w_pre, b_pre, g_pre, be_pre, m_pre, v_pre, preA, sq, out);
  knn_kernel<<<NB * (NPTS / 16), 256, 0, stream>>>(x, sq, nbr);
  mlp_kernel<<<NB * NPTS / 8, 256, 0, stream>>>(
      preA, nbr, w1, g1, be1, m1, v1, w2, g2, be2, m2, v2, w3, out);
}